// LDASAGE_68539088109979
// MI455X (gfx1250) — compile-verified
//
#include <hip/hip_runtime.h>
#include <math.h>

typedef float v2f __attribute__((ext_vector_type(2)));
typedef float v8f __attribute__((ext_vector_type(8)));

#define B_ 256
#define P_ 8
#define R_ 4
#define V_ 50000

#define LOG2E 1.44269504088896340736f   // 1/ln(2)
#define LN2   0.69314718055994530942f

__device__ __forceinline__ float fast_exp(float x) {
  return __builtin_amdgcn_exp2f(x * LOG2E);   // v_exp_f32
}
__device__ __forceinline__ float fast_log(float x) {
  return __builtin_amdgcn_logf(x) * LN2;      // v_log_f32
}

// ---------------------------------------------------------------------------
// Kernel A: Z[r*8+p] = sum_v exp(bg[v] + eta[p,v] + tau[r,v])   (32 blocks)
// ---------------------------------------------------------------------------
__global__ __launch_bounds__(256) void lda_z_kernel(
    const float* __restrict__ eta, const float* __restrict__ tau,
    const float* __restrict__ bg, float* __restrict__ Z) {
  __shared__ float red[256];
  const int rp = blockIdx.x;            // 0..31 -> (r, p)
  const int r = rp >> 3, p = rp & 7;
  const float* ep = eta + p * V_;
  const float* tr = tau + r * V_;
  float s = 0.f;
  for (int v = threadIdx.x; v < V_; v += 256)
    s += fast_exp(bg[v] + ep[v] + tr[v]);
  red[threadIdx.x] = s;
  __syncthreads();
  for (int off = 128; off > 0; off >>= 1) {
    if (threadIdx.x < off) red[threadIdx.x] += red[threadIdx.x + off];
    __syncthreads();
  }
  if (threadIdx.x == 0) Z[rp] = red[0];
}

// ---------------------------------------------------------------------------
// Kernel B: coef[b,p] = softmax(theta_logits[char(b)])[p] / Z[role(b),p]
// ---------------------------------------------------------------------------
__global__ __launch_bounds__(256) void lda_coef_kernel(
    const int* __restrict__ char_idx, const int* __restrict__ role_idx,
    const float* __restrict__ theta_logits, const float* __restrict__ Z,
    float* __restrict__ coef) {
  const int b = threadIdx.x;            // one thread per batch row
  const int c = char_idx[b];
  const int r = role_idx[b];
  float l[P_];
  float mx = -1e30f;
#pragma unroll
  for (int k = 0; k < P_; ++k) { l[k] = theta_logits[c * P_ + k]; mx = fmaxf(mx, l[k]); }
  float s = 0.f;
#pragma unroll
  for (int k = 0; k < P_; ++k) { l[k] = fast_exp(l[k] - mx); s += l[k]; }
  const float inv = __builtin_amdgcn_rcpf(s);
#pragma unroll
  for (int k = 0; k < P_; ++k)
    coef[b * P_ + k] = l[k] * inv * __builtin_amdgcn_rcpf(Z[r * P_ + k]);
}

// ---------------------------------------------------------------------------
// Kernel C: out[b,v] = log( exp(bg[v]+tau[r(b),v]) * (coef[b,:]·E[:,v]) + 1e-10 )
// GEMM M=256, N=V, K=8 via V_WMMA_F32_16X16X4_F32 (two chained K=4 steps).
// Block = 256 threads = 8 waves; wave w owns a 16-column V tile; loops over
// 16 M-tiles covering all 256 batch rows. coef + role selectors in LDS.
// Epilogue is fully branchless: role select via two fma selectors, raw
// v_log_f32 for the final log (argument is always normal, >= 1e-10).
// ---------------------------------------------------------------------------
__global__ __launch_bounds__(256) void lda_mix_kernel(
    const int* __restrict__ role_idx, const float* __restrict__ eta,
    const float* __restrict__ tau, const float* __restrict__ bg,
    const float* __restrict__ coef, float* __restrict__ out) {
  __shared__ float  s_coef[B_ * P_];    // 8 KB
  __shared__ float2 s_rm[B_];           // 2 KB: (role&1, role>>1) as floats

  const int t = threadIdx.x;
#pragma unroll
  for (int k = 0; k < P_; ++k) s_coef[t * P_ + k] = coef[t * P_ + k];
  {
    const int role = role_idx[t];
    s_rm[t] = make_float2((float)(role & 1), (float)(role >> 1));
  }
  __syncthreads();

  const int wave = t >> 5;
  const int lane = t & 31;
  const int half = lane >> 4;           // 0: lanes 0-15, 1: lanes 16-31
  const int lc   = lane & 15;           // column within tile / M within A half
  const int koff = half * 2;            // A/B fragment K sub-offset per ISA layout

  const int v0   = blockIdx.x * 128 + wave * 16;
  const int vcol = v0 + lc;
  const int vc   = vcol < V_ ? vcol : V_ - 1;   // clamp loads; stores predicated

  // B fragments (4x16 f32, K x N): E[p,v] = exp(eta[p,v])
  v2f bf0, bf1;
  bf0.x = fast_exp(eta[(0 + koff) * V_ + vc]);  // K = 0 / 2
  bf0.y = fast_exp(eta[(1 + koff) * V_ + vc]);  // K = 1 / 3
  bf1.x = fast_exp(eta[(4 + koff) * V_ + vc]);  // K = 4 / 6
  bf1.y = fast_exp(eta[(5 + koff) * V_ + vc]);  // K = 5 / 7

  // Per-role output scale exp(bg+tau) for this column (R = 4)
  const float bgv = bg[vc];
  const float e0 = fast_exp(bgv + tau[0 * V_ + vc]);
  const float e1 = fast_exp(bgv + tau[1 * V_ + vc]);
  const float e2 = fast_exp(bgv + tau[2 * V_ + vc]);
  const float e3 = fast_exp(bgv + tau[3 * V_ + vc]);
  const float d10 = e1 - e0;            // selector deltas (branchless pick)
  const float d32 = e3 - e2;

  for (int tile = 0; tile < 16; ++tile) {
    const int m0 = tile * 16;
    // A fragment (16x4 f32): lanes 0-15 hold K0/K1, lanes 16-31 hold K2/K3
    const float* crow = &s_coef[(m0 + lc) * P_ + koff];
    v2f af0 = { crow[0], crow[1] };               // K base 0
    v2f af1 = { crow[4], crow[5] };               // K base 4
    v8f acc = {};
    acc = __builtin_amdgcn_wmma_f32_16x16x4_f32(false, af0, false, bf0,
                                                (short)0, acc, false, false);
    acc = __builtin_amdgcn_wmma_f32_16x16x4_f32(false, af1, false, bf1,
                                                (short)0, acc, false, false);
    if (vcol < V_) {
#pragma unroll
      for (int rr = 0; rr < 8; ++rr) {
        const int m = m0 + rr + half * 8;         // C/D layout: rows r and r+8
        const float2 rm = s_rm[m];                // one ds_load_b64
        // branchless 4-way select of exp(bg+tau[role,m-col]):
        const float elo = fmaf(rm.x, d10, e0);    // role in {0,1}
        const float ehi = fmaf(rm.x, d32, e2);    // role in {2,3}
        const float ebt = fmaf(rm.y, ehi - elo, elo);
        const float prob = fmaf(ebt, acc[rr], 1e-10f);
        out[(size_t)m * V_ + vcol] = fast_log(prob);
      }
    }
  }
}

// ---------------------------------------------------------------------------
extern "C" void kernel_launch(void* const* d_in, const int* in_sizes, int n_in,
                              void* d_out, int out_size, void* d_ws, size_t ws_size,
                              hipStream_t stream) {
  const int*   char_idx     = (const int*)d_in[0];
  const int*   role_idx     = (const int*)d_in[1];
  const float* theta_logits = (const float*)d_in[2];
  const float* eta          = (const float*)d_in[3];
  const float* tau          = (const float*)d_in[4];
  const float* bg           = (const float*)d_in[5];
  float*       out          = (float*)d_out;

  float* Z    = (float*)d_ws;        // 32 floats
  float* coef = Z + 32;              // 256*8 floats

  lda_z_kernel<<<R_ * P_, 256, 0, stream>>>(eta, tau, bg, Z);
  lda_coef_kernel<<<1, 256, 0, stream>>>(char_idx, role_idx, theta_logits, Z, coef);
  lda_mix_kernel<<<(V_ + 127) / 128, 256, 0, stream>>>(role_idx, eta, tau, bg,
                                                       coef, out);
}